// ParabolicPool2D_SS_40106404610420
// MI455X (gfx1250) — compile-verified
//
#include <hip/hip_runtime.h>
#include <hip/hip_bf16.h>
#include <stdint.h>

// Problem constants (from reference)
#define KS      7
#define PADK    3
#define STRIDE  2
#define BATCH   16
#define CHAN    128
#define H       256
#define W       256
#define HO      128
#define WO      128

// Tiling
#define TO      16                         // output rows per block
#define IN_ROWS (STRIDE * TO + KS - 2)     // 37 input rows per strip
#define FW      264                        // LDS row pitch in floats (264*4 = 1056 B, 16B multiple)
#define FOFF    4                          // data starts at float index 4 (16B aligned), col c -> idx c+4

typedef float v4f __attribute__((ext_vector_type(4)));
typedef int   v4i __attribute__((ext_vector_type(4)));
// pointer-to-AS1 v4i (clang/HIP prints AS1 as __device__) and pointer-to-AS3 v4i (__shared__)
typedef __attribute__((address_space(1))) v4i* gv4p;
typedef __attribute__((address_space(3))) v4i* lv4p;

__global__ __launch_bounds__(256)
void ParabolicPool2D_kernel(const float* __restrict__ f,
                            const float* __restrict__ tptr,
                            float* __restrict__ out)
{
    __shared__ float sf[IN_ROWS * FW];   // staged input rows (+/- inf apron)
    __shared__ float sg[IN_ROWS * WO];   // horizontally dilated intermediate

    const int tid   = threadIdx.x;
    const int strip = blockIdx.x;        // 0..7
    const int plane = blockIdx.y;        // 0..2047  (b*C + c)

    const int  i0 = strip * TO;                 // first output row of this strip
    const int  r0 = STRIDE * i0 - PADK;         // first input row (may be negative)
    const long planeBaseIn  = (long)plane * H  * W;
    const long planeBaseOut = (long)plane * HO * WO;
    const float NEG_INF = -__builtin_inff();

    // ---- 1) issue async DMA for 37 rows x 1024B ASAP (apron fill overlaps it) ----
    // 256 threads -> 4 rows per issue (64 lanes x 16B per 256-float row)
    for (int rb = 0; rb < IN_ROWS; rb += 4) {
        const int rl = rb + (tid >> 6);         // local row
        const int ck = (tid & 63);              // 16-byte chunk within row
        if (rl < IN_ROWS) {
            const int rg = r0 + rl;             // global input row
            float* ldst = &sf[rl * FW + FOFF + ck * 4];
            if ((unsigned)rg < (unsigned)H) {
                const float* gsrc = f + planeBaseIn + (long)rg * W + ck * 4;
#if __has_builtin(__builtin_amdgcn_global_load_async_to_lds_b128)
                __builtin_amdgcn_global_load_async_to_lds_b128(
                    (gv4p)(uintptr_t)gsrc,
                    (lv4p)(uintptr_t)ldst,
                    0, 0);
#else
                *(v4f*)ldst = *(const v4f*)gsrc;
#endif
            } else {
                v4f m; m[0] = m[1] = m[2] = m[3] = NEG_INF;   // rows outside the image
                *(v4f*)ldst = m;
            }
        }
    }

    // ---- 2) -inf apron (disjoint from DMA range: indices 0..3 and 260..263) ----
    for (int e = tid; e < IN_ROWS * 8; e += 256) {
        const int r = e >> 3;
        const int c = e & 7;
        const int idx = (c < 4) ? c : (W + FOFF + (c - 4));
        sf[r * FW + idx] = NEG_INF;
    }

    // separable parabolic taps: hk[v] = -(v-3)^2 / (4t), identical for u and v
    const float t = tptr[0];
    const float s = -1.0f / (4.0f * t);
    float hk[KS];
#pragma unroll
    for (int v = 0; v < KS; ++v) {
        const float z = (float)(v - PADK);
        hk[v] = z * z * s;
    }

#if __has_builtin(__builtin_amdgcn_s_wait_asynccnt)
    __builtin_amdgcn_s_wait_asynccnt(0);
#else
    asm volatile("s_wait_asynccnt 0" ::: "memory");
#endif
    __syncthreads();

    // ---- 3) horizontal max-plus dilation: sg[r][j] = max_v sf[r][2j+v-3] + hk[v] ----
    // LDS idx for col (2j+v-3) is 2j+v+1; stride-2 lane access -> conflict-free on 64 banks.
    for (int e = tid; e < IN_ROWS * WO; e += 256) {
        const int r = e >> 7;                   // /128
        const int j = e & 127;
        const float* row = &sf[r * FW + 2 * j + 1];
        float m = row[0] + hk[0];
#pragma unroll
        for (int v = 1; v < KS; ++v)
            m = fmaxf(m, row[v] + hk[v]);
        sg[r * WO + j] = m;
    }
    __syncthreads();

    // ---- 4) vertical max-plus, rolling 7-row register window (shift by 2/row) ----
    // Thread -> fixed column j, 8 consecutive output rows. Consecutive rows share
    // 5/7 g-rows: 7 initial + 2/step LDS loads instead of 7/step.
    {
        const int j     = tid & 127;            // column
        const int tbase = (tid >> 7) * 8;       // 0 or 8: first output row in this thread's run
        const float* gcol = &sg[(2 * tbase) * WO + j];

        float w0 = gcol[0 * WO], w1 = gcol[1 * WO], w2 = gcol[2 * WO],
              w3 = gcol[3 * WO], w4 = gcol[4 * WO], w5 = gcol[5 * WO],
              w6 = gcol[6 * WO];

#pragma unroll
        for (int ti = 0; ti < 8; ++ti) {
            float m = w0 + hk[0];
            m = fmaxf(m, w1 + hk[1]);
            m = fmaxf(m, w2 + hk[2]);
            m = fmaxf(m, w3 + hk[3]);
            m = fmaxf(m, w4 + hk[4]);
            m = fmaxf(m, w5 + hk[5]);
            m = fmaxf(m, w6 + hk[6]);
            __builtin_nontemporal_store(
                m, &out[planeBaseOut + (long)(i0 + tbase + ti) * WO + j]);
            if (ti < 7) {                       // slide window down by 2 input rows
                w0 = w2; w1 = w3; w2 = w4; w3 = w5; w4 = w6;
                w5 = gcol[(2 * ti + 7) * WO];
                w6 = gcol[(2 * ti + 8) * WO];
            }
        }
    }
}

extern "C" void kernel_launch(void* const* d_in, const int* in_sizes, int n_in,
                              void* d_out, int out_size, void* d_ws, size_t ws_size,
                              hipStream_t stream) {
    (void)in_sizes; (void)n_in; (void)out_size; (void)d_ws; (void)ws_size;
    const float* f = (const float*)d_in[0];
    const float* t = (const float*)d_in[1];
    float* out     = (float*)d_out;

    dim3 grid(HO / TO, BATCH * CHAN);   // (8 strips, 2048 planes)
    ParabolicPool2D_kernel<<<grid, 256, 0, stream>>>(f, t, out);
}